// PoolingLayer_85744727097809
// MI455X (gfx1250) — compile-verified
//
#include <hip/hip_runtime.h>
#include <cstdint>
#include <cstddef>

// ---------------------------------------------------------------------------
// e3nn FullyConnectedTensorProduct + o3.Linear on irreps 32x0e+32x1o+16x2e,
// fused into f32 WMMA GEMMs (V_WMMA_F32_16X16X4_F32) on gfx1250.
//   out[z, w, k] = sum_paths pw * sum_uv W[uv,w] * t[z,uv,k] + Linear + bias
// One wave32 handles 32 rows (two 16-row M-tiles) so every B fragment
// (4x16 slice of W) feeds two WMMAs per k -> half the L2 weight traffic.
// ---------------------------------------------------------------------------

typedef float v2f __attribute__((ext_vector_type(2)));
typedef float v8f __attribute__((ext_vector_type(8)));

struct Coef { int i, j, k; float v; };

// --- real Wigner-3j tables (Frobenius norm 1, e3nn index order y,z,x) ------
__device__ constexpr float S3 = 0.57735026918962576f;   // 1/sqrt(3)
__device__ constexpr float S5 = 0.44721359549995794f;   // 1/sqrt(5)
__device__ constexpr float T2C = 0.31622776601683794f;  // 1/sqrt(10)
__device__ constexpr float T2D = 0.36514837167011072f;  // 2/sqrt(30)
__device__ constexpr float T2E = 0.18257418583505536f;  // 1/sqrt(30)
__device__ constexpr float A222 = 0.2390457218668787f;  // sqrt(96/35)/(4*sqrt(3))
__device__ constexpr float B222 = 0.1195228609334393f;  // A222/2
__device__ constexpr float G222 = 0.2070196678027062f;  // sqrt(96/35)/8

__device__ constexpr Coef C000[1]  = {{0,0,0,1.0f}};
__device__ constexpr Coef C011[3]  = {{0,0,0,S3},{0,1,1,S3},{0,2,2,S3}};
__device__ constexpr Coef C101[3]  = {{0,0,0,S3},{1,0,1,S3},{2,0,2,S3}};
__device__ constexpr Coef C110[3]  = {{0,0,0,S3},{1,1,0,S3},{2,2,0,S3}};
__device__ constexpr Coef C022[5]  = {{0,0,0,S5},{0,1,1,S5},{0,2,2,S5},{0,3,3,S5},{0,4,4,S5}};
__device__ constexpr Coef C202[5]  = {{0,0,0,S5},{1,0,1,S5},{2,0,2,S5},{3,0,3,S5},{4,0,4,S5}};
__device__ constexpr Coef C220[5]  = {{0,0,0,S5},{1,1,0,S5},{2,2,0,S5},{3,3,0,S5},{4,4,0,S5}};
__device__ constexpr Coef C112[11] = {
  {2,0,0,T2C},{0,2,0,T2C},{0,1,1,T2C},{1,0,1,T2C},
  {1,1,2,T2D},{0,0,2,-T2E},{2,2,2,-T2E},
  {2,1,3,T2C},{1,2,3,T2C},{2,2,4,T2C},{0,0,4,-T2C}};
__device__ constexpr Coef C121[11] = {   // C121[i,k,j] = C112[i,j,k]
  {2,0,0,T2C},{0,0,2,T2C},{0,1,1,T2C},{1,1,0,T2C},
  {1,2,1,T2D},{0,2,0,-T2E},{2,2,2,-T2E},
  {2,3,1,T2C},{1,3,2,T2C},{2,4,2,T2C},{0,4,0,-T2C}};
__device__ constexpr Coef C211[11] = {   // C211[k,i,j] = C112[i,j,k]
  {0,2,0,T2C},{0,0,2,T2C},{1,0,1,T2C},{1,1,0,T2C},
  {2,1,1,T2D},{2,0,0,-T2E},{2,2,2,-T2E},
  {3,2,1,T2C},{3,1,2,T2C},{4,2,2,T2C},{4,0,0,-T2C}};
__device__ constexpr Coef C222[25] = {
  {2,2,2, A222},
  {2,0,0,-A222},{0,2,0,-A222},{0,0,2,-A222},
  {2,4,4,-A222},{4,2,4,-A222},{4,4,2,-A222},
  {2,1,1, B222},{1,2,1, B222},{1,1,2, B222},
  {2,3,3, B222},{3,2,3, B222},{3,3,2, B222},
  {4,1,1,-G222},{1,4,1,-G222},{1,1,4,-G222},
  {4,3,3, G222},{3,4,3, G222},{3,3,4, G222},
  {0,1,3, G222},{0,3,1, G222},{1,0,3, G222},{1,3,0, G222},{3,0,1, G222},{3,1,0, G222}};

// path normalizations pw = sqrt((2lo+1)/fan_in(io))
__device__ constexpr float PW0 = 0.0208333333333333f;   // 1/48        (io=0e)
__device__ constexpr float PW1 = 0.03125f;              // 1/32        (io=1o)
__device__ constexpr float PW2 = 0.0465847495312456f;   // sqrt(5)/48  (io=2e)
__device__ constexpr float LS32 = 0.1767766952966369f;  // 1/sqrt(32)
__device__ constexpr float LS16 = 0.25f;                // 1/sqrt(16)

#define ROWS 32       // rows per wave (two 16-row WMMA M-tiles)
#define LSTRIDE 212   // 208 + pad -> 16 consecutive rows hit distinct LDS banks

__device__ inline v8f vzero8() {
  v8f z;
#pragma unroll
  for (int e = 0; e < 8; ++e) z[e] = 0.0f;
  return z;
}

// D = A(16x4 f32) x B(4x16 f32) + C   -- wave32 fp32 matrix op
__device__ inline v8f wmma4(v2f a, v2f b, v8f c) {
  return __builtin_amdgcn_wmma_f32_16x16x4_f32(false, a, false, b, (short)0, c,
                                               false, false);
}

// One tensor-product path, two M-tiles sharing each B fragment.
// Lane layout per ISA 7.12.2: A v0 <-> K=kk, v1 <-> K=kk+1 (kk = 0 | 2);
// B v0/v1 analogous with N = lane&15.
template<int M1,int L1,int OFF1,int M2,int L2,int OFF2,int MO,int LO,int NNZ>
__device__ inline void tp_path(const Coef (&CF)[NNZ],
                               const float* __restrict__ wb, float pw,
                               const float* __restrict__ xrowA,
                               const float* __restrict__ xrowB,
                               int kk, int ncol, v8f* accA, v8f* accB) {
  constexpr int D1 = 2*L1+1, D2 = 2*L2+1, DO = 2*LO+1, NW = MO/16;
  for (int q = 0; q < (M1*M2)/4; ++q) {
    const int uv = 4*q + kk;          // kk in {0,2}; uv,uv+1 share the same u
    const int u  = uv / M2;
    const int v  = uv % M2;
    float x1A[D1], x1B[D1], x2aA[D2], x2bA[D2], x2aB[D2], x2bB[D2];
#pragma unroll
    for (int i = 0; i < D1; ++i) {
      x1A[i] = xrowA[OFF1 + u*D1 + i];
      x1B[i] = xrowB[OFF1 + u*D1 + i];
    }
#pragma unroll
    for (int j = 0; j < D2; ++j) {
      x2aA[j] = xrowA[OFF2 + v*D2 + j];
      x2bA[j] = xrowA[OFF2 + (v+1)*D2 + j];
      x2aB[j] = xrowB[OFF2 + v*D2 + j];
      x2bB[j] = xrowB[OFF2 + (v+1)*D2 + j];
    }
    v2f A0[DO], A1[DO];
#pragma unroll
    for (int k = 0; k < DO; ++k) {
      A0[k].x = 0.0f; A0[k].y = 0.0f;
      A1[k].x = 0.0f; A1[k].y = 0.0f;
    }
#pragma unroll
    for (int n = 0; n < NNZ; ++n) {   // constexpr-folded 3j contraction
      A0[CF[n].k].x += CF[n].v * x1A[CF[n].i] * x2aA[CF[n].j];
      A0[CF[n].k].y += CF[n].v * x1A[CF[n].i] * x2bA[CF[n].j];
      A1[CF[n].k].x += CF[n].v * x1B[CF[n].i] * x2aB[CF[n].j];
      A1[CF[n].k].y += CF[n].v * x1B[CF[n].i] * x2bB[CF[n].j];
    }
#pragma unroll
    for (int wt = 0; wt < NW; ++wt) {
      const float* wp = wb + (size_t)uv*MO + wt*16 + ncol;
      v2f B; B.x = pw * wp[0]; B.y = pw * wp[MO];   // K rows kk, kk+1
#pragma unroll
      for (int k = 0; k < DO; ++k) {                // B reused across k & tiles
        accA[k*NW + wt] = wmma4(A0[k], B, accA[k*NW + wt]);
        accB[k*NW + wt] = wmma4(A1[k], B, accB[k*NW + wt]);
      }
    }
  }
}

// o3.Linear block as an extra WMMA path: A = x columns, B = W/sqrt(m)
template<int M,int L,int OFF>
__device__ inline void lin_path(const float* __restrict__ wl, float scale,
                                const float* __restrict__ xrowA,
                                const float* __restrict__ xrowB,
                                int kk, int ncol, v8f* accA, v8f* accB) {
  constexpr int D = 2*L+1, NW = M/16;
  for (int q = 0; q < M/4; ++q) {
    const int ua = 4*q + kk;
    v2f A0[D], A1[D];
#pragma unroll
    for (int k = 0; k < D; ++k) {
      A0[k].x = xrowA[OFF + ua*D + k];
      A0[k].y = xrowA[OFF + (ua+1)*D + k];
      A1[k].x = xrowB[OFF + ua*D + k];
      A1[k].y = xrowB[OFF + (ua+1)*D + k];
    }
#pragma unroll
    for (int wt = 0; wt < NW; ++wt) {
      const float* wp = wl + (size_t)ua*M + wt*16 + ncol;
      v2f B; B.x = scale * wp[0]; B.y = scale * wp[M];
#pragma unroll
      for (int k = 0; k < D; ++k) {
        accA[k*NW + wt] = wmma4(A0[k], B, accA[k*NW + wt]);
        accB[k*NW + wt] = wmma4(A1[k], B, accB[k*NW + wt]);
      }
    }
  }
}

__global__ __launch_bounds__(64)
void eqtp_fused(const float* __restrict__ x,
                const float* __restrict__ tpw,
                const float* __restrict__ linw,
                const float* __restrict__ linb,
                float* __restrict__ out, int nrows) {
  __shared__ float xt[2][ROWS * LSTRIDE];   // 2 waves x 32 rows x 212 floats
  const int lane = threadIdx.x & 31;
  const int wv   = threadIdx.x >> 5;
  const int tile = blockIdx.x * 2 + wv;      // one wave == one 32-row tile
  if (tile * ROWS >= nrows) return;          // wave-uniform guard
  const int z0 = tile * ROWS;
  float* tl = xt[wv];

  // ---- stage the 32x208 activation tile into LDS (padded stride) ----------
  for (int t = lane; t < ROWS * 52; t += 32) {
    const int r = t / 52, c4 = t % 52;
    int zr = z0 + r; if (zr > nrows - 1) zr = nrows - 1;   // tail clamp
    const float4 vch = ((const float4*)(x + (size_t)zr * 208))[c4];
    *(float4*)(tl + r * LSTRIDE + c4 * 4) = vch;
  }
  asm volatile("s_wait_dscnt 0x0" ::: "memory");  // cross-lane LDS visibility

  const int ncol  = lane & 15;            // N index (also lane's M row in tile)
  const int kk    = (lane >> 4) << 1;     // K sub-index: 0 | 2
  const int rbase = (lane >> 4) * 8;      // C/D rows: r (lo half) / r+8 (hi)
  const float* xrowA = tl + ncol * LSTRIDE;          // rows z0 .. z0+15
  const float* xrowB = tl + (16 + ncol) * LSTRIDE;   // rows z0+16 .. z0+31

  // ============== output irrep 0: 32x0e  (acc[wt], wt<2) ===================
  {
    v8f aA[2], aB[2];
#pragma unroll
    for (int e = 0; e < 2; ++e) { aA[e] = vzero8(); aB[e] = vzero8(); }
    tp_path<32,0,0,  32,0,0,  32,0>(C000, tpw + 0,      PW0, xrowA, xrowB, kk, ncol, aA, aB);
    tp_path<32,1,32, 32,1,32, 32,0>(C110, tpw + 106496, PW0, xrowA, xrowB, kk, ncol, aA, aB);
    tp_path<16,2,128,16,2,128,32,0>(C220, tpw + 196608, PW0, xrowA, xrowB, kk, ncol, aA, aB);
    lin_path<32,0,0>(linw + 0, LS32, xrowA, xrowB, kk, ncol, aA, aB);
#pragma unroll
    for (int wt = 0; wt < 2; ++wt) {
      const float bv = linb[wt*16 + ncol];
#pragma unroll
      for (int r = 0; r < 8; ++r) {
        const int za = z0 + rbase + r, zb = za + 16;
        if (za < nrows) out[(size_t)za*208 + (wt*16 + ncol)] = aA[wt][r] + bv;
        if (zb < nrows) out[(size_t)zb*208 + (wt*16 + ncol)] = aB[wt][r] + bv;
      }
    }
  }

  // ============== output irrep 1: 32x1o  (acc[k*2+wt], k<3) ================
  {
    v8f aA[6], aB[6];
#pragma unroll
    for (int e = 0; e < 6; ++e) { aA[e] = vzero8(); aB[e] = vzero8(); }
    tp_path<32,0,0,  32,1,32, 32,1>(C011, tpw + 32768,  PW1, xrowA, xrowB, kk, ncol, aA, aB);
    tp_path<32,1,32, 32,0,0,  32,1>(C101, tpw + 73728,  PW1, xrowA, xrowB, kk, ncol, aA, aB);
    tp_path<32,1,32, 16,2,128,32,1>(C121, tpw + 155648, PW1, xrowA, xrowB, kk, ncol, aA, aB);
    tp_path<16,2,128,32,1,32, 32,1>(C211, tpw + 180224, PW1, xrowA, xrowB, kk, ncol, aA, aB);
    lin_path<32,1,32>(linw + 1024, LS32, xrowA, xrowB, kk, ncol, aA, aB);
#pragma unroll
    for (int k = 0; k < 3; ++k)
#pragma unroll
      for (int wt = 0; wt < 2; ++wt)
#pragma unroll
        for (int r = 0; r < 8; ++r) {
          const int za = z0 + rbase + r, zb = za + 16;
          const int col = 32 + (wt*16 + ncol)*3 + k;
          if (za < nrows) out[(size_t)za*208 + col] = aA[k*2 + wt][r];
          if (zb < nrows) out[(size_t)zb*208 + col] = aB[k*2 + wt][r];
        }
  }

  // ============== output irrep 2: 16x2e  (acc[k], k<5) =====================
  {
    v8f aA[5], aB[5];
#pragma unroll
    for (int e = 0; e < 5; ++e) { aA[e] = vzero8(); aB[e] = vzero8(); }
    tp_path<32,0,0,  16,2,128,16,2>(C022, tpw + 65536,  PW2, xrowA, xrowB, kk, ncol, aA, aB);
    tp_path<32,1,32, 32,1,32, 16,2>(C112, tpw + 139264, PW2, xrowA, xrowB, kk, ncol, aA, aB);
    tp_path<16,2,128,32,0,0,  16,2>(C202, tpw + 172032, PW2, xrowA, xrowB, kk, ncol, aA, aB);
    tp_path<16,2,128,16,2,128,16,2>(C222, tpw + 204800, PW2, xrowA, xrowB, kk, ncol, aA, aB);
    lin_path<16,2,128>(linw + 2048, LS16, xrowA, xrowB, kk, ncol, aA, aB);
#pragma unroll
    for (int k = 0; k < 5; ++k)
#pragma unroll
      for (int r = 0; r < 8; ++r) {
        const int za = z0 + rbase + r, zb = za + 16;
        const int col = 128 + ncol*5 + k;
        if (za < nrows) out[(size_t)za*208 + col] = aA[k][r];
        if (zb < nrows) out[(size_t)zb*208 + col] = aB[k][r];
      }
  }
}

extern "C" void kernel_launch(void* const* d_in, const int* in_sizes, int n_in,
                              void* d_out, int out_size, void* d_ws, size_t ws_size,
                              hipStream_t stream) {
  (void)n_in; (void)out_size; (void)d_ws; (void)ws_size;
  const float* node_x  = (const float*)d_in[0];
  const float* edge_x  = (const float*)d_in[1];
  const float* node_tp = (const float*)d_in[2];
  const float* edge_tp = (const float*)d_in[3];
  const float* node_lw = (const float*)d_in[4];
  const float* node_lb = (const float*)d_in[5];
  const float* edge_lw = (const float*)d_in[6];
  const float* edge_lb = (const float*)d_in[7];
  float* out = (float*)d_out;

  const int nn = in_sizes[0] / 208;   // 20000
  const int ne = in_sizes[1] / 208;   // 60000
  float* out_node = out;
  float* out_edge = out + (size_t)nn * 208;

  if (nn > 0) {
    const int tiles = (nn + ROWS - 1) / ROWS;
    eqtp_fused<<<(tiles + 1) / 2, 64, 0, stream>>>(node_x, node_tp, node_lw,
                                                   node_lb, out_node, nn);
  }
  if (ne > 0) {
    const int tiles = (ne + ROWS - 1) / ROWS;
    eqtp_fused<<<(tiles + 1) / 2, 64, 0, stream>>>(edge_x, edge_tp, edge_lw,
                                                   edge_lb, out_edge, ne);
  }
}